// MSDN_block_73005854098149
// MI455X (gfx1250) — compile-verified
//
#include <hip/hip_runtime.h>

#define NB   4
#define WD   256
#define HW   (WD * WD)      // 65536
#define NCH  51
#define NPIX (NB * HW)      // 262144
#define EPSV 1e-5f
#define MAXD 10

typedef __attribute__((ext_vector_type(2))) float v2f;
typedef __attribute__((ext_vector_type(8))) float v8f;

#if defined(__has_builtin)
#if __has_builtin(__builtin_amdgcn_wmma_f32_16x16x4_f32)
#define HAVE_WMMA_F32X4 1
#endif
#if __has_builtin(__builtin_amdgcn_s_wait_asynccnt)
#define WAIT_ASYNC() __builtin_amdgcn_s_wait_asynccnt(0)
#endif
#endif
#ifndef HAVE_WMMA_F32X4
#define HAVE_WMMA_F32X4 0
#endif
#ifndef WAIT_ASYNC
#define WAIT_ASYNC() asm volatile("s_wait_asynccnt 0x0" ::: "memory")
#endif

// CDNA5 async copy: memory -> LDS directly through the VMEM pipe (ASYNCcnt).
// LDS byte address = low 32 bits of the flat pointer.
__device__ __forceinline__ void async_copy_b32(void* lds_dst, const float* gsrc) {
    unsigned int l = (unsigned int)(size_t)lds_dst;
    asm volatile("global_load_async_to_lds_b32 %0, %1, off"
                 :: "v"(l), "v"(gsrc)
                 : "memory");
}

// ---------------- deterministic block-wide (sum, sumsq) reduction ----------------
__device__ __forceinline__ float2 block_pair_reduce(float s, float q) {
    __shared__ float rs[256];
    __shared__ float rq[256];
    const int t = threadIdx.x;
    __syncthreads();            // WAR guard if helper is reused in one kernel
    rs[t] = s; rq[t] = q;
    __syncthreads();
    #pragma unroll
    for (int o = 128; o > 0; o >>= 1) {
        if (t < o) { rs[t] += rs[t + o]; rq[t] += rq[t + o]; }
        __syncthreads();
    }
    return make_float2(rs[0], rq[0]);
}

// every thread returns the full (sum, sumsq) of npairs block-partials
__device__ __forceinline__ float2 reduce_pairs(const float* part, int npairs) {
    float s = 0.f, q = 0.f;
    for (int i = threadIdx.x; i < npairs; i += 256) { s += part[2 * i]; q += part[2 * i + 1]; }
    return block_pair_reduce(s, q);
}

// ---------------- stage 1: stats of x ----------------
__global__ void xstats_kernel(const float* __restrict__ x, float* __restrict__ xpart, int n) {
    float s = 0.f, q = 0.f;
    for (int i = blockIdx.x * 256 + threadIdx.x; i < n; i += 256 * 256) {
        float v = x[i]; s += v; q += v * v;
    }
    float2 r = block_pair_reduce(s, q);
    if (threadIdx.x == 0) { xpart[2 * blockIdx.x] = r.x; xpart[2 * blockIdx.x + 1] = r.y; }
}

// ---------------- stage 2: h = conv1x1(relu(bn1(x))) -> feats ch 0 ----------------
__global__ void layer0_kernel(const float* __restrict__ x, float* __restrict__ feats,
                              const float* __restrict__ xpart,
                              const float* __restrict__ g, const float* __restrict__ bb,
                              const float* __restrict__ cw, const float* __restrict__ cbias,
                              float* __restrict__ zpart) {
    float2 sv = reduce_pairs(xpart, 256);
    float m = sv.x * (1.f / NPIX);
    float v = sv.y * (1.f / NPIX) - m * m;
    float S = g[0] * rsqrtf(v + EPSV);
    float T = bb[0] - m * S;
    float w0 = cw[0], c0 = cbias[0];
    int idx = blockIdx.x * 256 + threadIdx.x;
    int b = idx >> 16, p = idx & (HW - 1);
    float h = fmaf(fmaxf(fmaf(x[idx], S, T), 0.f), w0, c0);
    feats[(size_t)b * NCH * HW + p] = h;
    float2 r = block_pair_reduce(h, h * h);
    if (threadIdx.x == 0) { zpart[2 * blockIdx.x] = r.x; zpart[2 * blockIdx.x + 1] = r.y; }
}

// ---------------- per-layer dilated 3x3 conv (cin -> 1), pre-BN, + y stats ----------------
// one block per (batch, output row); async-copies 3 haloed input rows per channel into LDS
__global__ void conv_layer_kernel(float* __restrict__ feats,
                                  const float* __restrict__ w,
                                  const float* __restrict__ cb,
                                  float* __restrict__ ypart,
                                  int cin, int dil) {
    __shared__ float wl[NCH * 9];
    __shared__ float rows[3][WD + 2 * MAXD];
    const int t = threadIdx.x;
    const int b = blockIdx.x >> 8;
    const int h = blockIdx.x & (WD - 1);
    for (int k = t; k < cin * 9; k += 256) wl[k] = w[k];
    float* fb = feats + (size_t)b * NCH * HW;
    float acc = cb[0];
    const int span = WD + 2 * dil;
    for (int c = 0; c < cin; ++c) {
        const float* ch = fb + (size_t)c * HW;
        __syncthreads();                       // rows reuse guard (+ wl ready on iter 0)
        #pragma unroll
        for (int r = 0; r < 3; ++r) {
            int sh = h + (r - 1) * dil;
            bool okr = (sh >= 0) && (sh < WD);
            const float* rp = ch + (size_t)sh * WD;
            for (int idx = t; idx < span; idx += 256) {
                int col = idx - dil;
                if (okr && col >= 0 && col < WD) {
                    async_copy_b32(&rows[r][idx], rp + col);   // ASYNCcnt path
                } else {
                    rows[r][idx] = 0.f;                        // zero-pad halo (DScnt)
                }
            }
        }
        WAIT_ASYNC();                          // this wave's async LDS writes landed
        __syncthreads();                       // all waves' writes visible
        if (c + 1 < cin) __builtin_prefetch(ch + HW + (size_t)h * WD + t, 0, 0);
        const float* wc = &wl[c * 9];
        #pragma unroll
        for (int r = 0; r < 3; ++r) {
            acc = fmaf(wc[r * 3 + 0], rows[r][t],            acc);
            acc = fmaf(wc[r * 3 + 1], rows[r][t + dil],      acc);
            acc = fmaf(wc[r * 3 + 2], rows[r][t + 2 * dil],  acc);
        }
    }
    // pre-BN y written into the new feats channel (normalized in place next kernel)
    fb[(size_t)cin * HW + (size_t)h * WD + t] = acc;
    float2 r = block_pair_reduce(acc, acc * acc);
    if (t == 0) { ypart[2 * blockIdx.x] = r.x; ypart[2 * blockIdx.x + 1] = r.y; }
}

// ---------------- per-layer BN + ReLU in place, + post-activation stats ----------------
__global__ void bn_apply_kernel(float* __restrict__ feats, int c,
                                const float* __restrict__ ypart,
                                const float* __restrict__ g, const float* __restrict__ be,
                                float* __restrict__ zpart) {
    float2 sv = reduce_pairs(ypart, NB * WD);
    float m = sv.x * (1.f / NPIX);
    float v = sv.y * (1.f / NPIX) - m * m;
    float S = g[0] * rsqrtf(v + EPSV);
    float T = be[0] - m * S;
    int idx = blockIdx.x * 256 + threadIdx.x;
    int b = idx >> 16, p = idx & (HW - 1);
    float* ptr = feats + (size_t)b * NCH * HW + (size_t)c * HW + p;
    float z = fmaxf(fmaf(*ptr, S, T), 0.f);
    *ptr = z;
    float2 r = block_pair_reduce(z, z * z);
    if (threadIdx.x == 0) { zpart[2 * blockIdx.x] = r.x; zpart[2 * blockIdx.x + 1] = r.y; }
}

__global__ void finalize_kernel(const float* __restrict__ zpart, int npairs,
                                float* __restrict__ dst2) {
    float2 r = reduce_pairs(zpart, npairs);
    if (threadIdx.x == 0) { dst2[0] = r.x; dst2[1] = r.y; }
}

// ---------------- bn2 affine precompute: z = relu(feats*S_c + T_c) ----------------
__global__ void bn2_prep_kernel(const float* __restrict__ fstats,
                                const float* __restrict__ g,
                                const float* __restrict__ bb,
                                float* __restrict__ ST) {
    int c = threadIdx.x;
    if (c < NCH) {
        float m = fstats[2 * c] * (1.f / NPIX);
        float v = fstats[2 * c + 1] * (1.f / NPIX) - m * m;
        float S = g[c] * rsqrtf(v + EPSV);
        ST[c] = S;
        ST[NCH + c] = bb[c] - m * S;
    }
}

// ---------------- final fused bn2+relu+conv1x1 via V_WMMA_F32_16X16X4_F32 ----------------
// per wave: M = 16 pixels, K = 52 (51 ch + pad), N = 16 (out channels 0..2 used)
__global__ void final_conv_kernel(const float* __restrict__ feats,
                                  const float* __restrict__ ST,
                                  const float* __restrict__ w2,
                                  const float* __restrict__ b2,
                                  float* __restrict__ out) {
#if HAVE_WMMA_F32X4
    const int lane   = threadIdx.x & 31;
    const int wave   = (blockIdx.x * blockDim.x + threadIdx.x) >> 5;
    const int b      = wave >> 12;              // 4096 16-pixel tiles per batch image
    const int pixbase = (wave & 4095) << 4;
    const int mrow   = lane & 15;               // A row (pixel) == B column (out ch)
    const int khalf  = lane >> 4;               // K-half per ISA A/B layout
    const float* fb  = feats + (size_t)b * NCH * HW + pixbase + mrow;
    v8f acc = {};
    #pragma unroll
    for (int k0 = 0; k0 < 52; k0 += 4) {
        const int c0 = k0 + khalf * 2;
        const int c1 = c0 + 1;
        const int cc0 = (c0 < NCH) ? c0 : 0;
        const int cc1 = (c1 < NCH) ? c1 : 0;
        float a0 = fmaxf(fmaf(fb[(size_t)cc0 * HW], ST[cc0], ST[NCH + cc0]), 0.f);
        float a1 = fmaxf(fmaf(fb[(size_t)cc1 * HW], ST[cc1], ST[NCH + cc1]), 0.f);
        a0 = (c0 < NCH) ? a0 : 0.f;
        a1 = (c1 < NCH) ? a1 : 0.f;
        float bv0 = 0.f, bv1 = 0.f;
        if (mrow < 3) {
            bv0 = (c0 < NCH) ? w2[mrow * NCH + cc0] : 0.f;
            bv1 = (c1 < NCH) ? w2[mrow * NCH + cc1] : 0.f;
        }
        v2f A; A.x = a0;  A.y = a1;     // A[m=mrow, k], lanes0-15: k0,k0+1; lanes16-31: k0+2,k0+3
        v2f B; B.x = bv0; B.y = bv1;    // B[k, n=mrow], same K striping
        acc = __builtin_amdgcn_wmma_f32_16x16x4_f32(false, A, false, B,
                                                    (short)0, acc, false, false);
    }
    if (mrow < 3) {                     // D: VGPR r -> M = r + 8*khalf, N = mrow
        const float bias = b2[mrow];
        float* op = out + (size_t)b * 3 * HW + (size_t)mrow * HW + pixbase;
        #pragma unroll
        for (int r = 0; r < 8; ++r) op[r + 8 * khalf] = acc[r] + bias;
    }
#else
    int idx = blockIdx.x * blockDim.x + threadIdx.x;
    if (idx >= NPIX) return;
    int b = idx >> 16, p = idx & (HW - 1);
    const float* fb = feats + (size_t)b * NCH * HW + p;
    float o0 = b2[0], o1 = b2[1], o2 = b2[2];
    for (int c = 0; c < NCH; ++c) {
        float z = fmaxf(fmaf(fb[(size_t)c * HW], ST[c], ST[NCH + c]), 0.f);
        o0 = fmaf(w2[0 * NCH + c], z, o0);
        o1 = fmaf(w2[1 * NCH + c], z, o1);
        o2 = fmaf(w2[2 * NCH + c], z, o2);
    }
    out[(size_t)b * 3 * HW + 0 * HW + p] = o0;
    out[(size_t)b * 3 * HW + 1 * HW + p] = o1;
    out[(size_t)b * 3 * HW + 2 * HW + p] = o2;
#endif
}

// ---------------- host ----------------
extern "C" void kernel_launch(void* const* d_in, const int* in_sizes, int n_in,
                              void* d_out, int out_size, void* d_ws, size_t ws_size,
                              hipStream_t stream) {
    (void)in_sizes; (void)out_size; (void)ws_size;
    const float* x     = (const float*)d_in[0];
    const float* bn1_g = (const float*)d_in[1];
    const float* bn1_b = (const float*)d_in[2];
    const float* c1w   = (const float*)d_in[3];
    const float* c1b   = (const float*)d_in[4];
    // lists either expand to one input each (n_in == 209) or arrive concatenated (n_in == 13)
    const bool sep = (n_in > 13);
    const float* lw0 = (const float*)d_in[5];
    const float* lb0 = (const float*)d_in[6 % n_in];
    const float* lg0 = (const float*)d_in[7 % n_in];
    const float* lt0 = (const float*)d_in[8 % n_in];
    const float* bn2_g = (const float*)d_in[sep ? 205 : 9];
    const float* bn2_b = (const float*)d_in[sep ? 206 : 10];
    const float* c2w   = (const float*)d_in[sep ? 207 : 11];
    const float* c2b   = (const float*)d_in[sep ? 208 : 12];

    float* out   = (float*)d_out;
    float* feats = out + (size_t)3 * NPIX;      // second tuple output, written in place

    float* ws     = (float*)d_ws;
    float* xpart  = ws;                         // 512 floats  (256 blocks x pair)
    float* ypart  = ws + 512;                   // 2048 floats (1024 blocks x pair)
    float* zpart  = ws + 2560;                  // 2048 floats
    float* fstats = ws + 4608;                  // 102 floats  (51 channels x pair)
    float* ST     = ws + 4710;                  // 102 floats  (scale, shift)

    xstats_kernel<<<256, 256, 0, stream>>>(x, xpart, NPIX);
    layer0_kernel<<<NPIX / 256, 256, 0, stream>>>(x, feats, xpart, bn1_g, bn1_b, c1w, c1b, zpart);
    finalize_kernel<<<1, 256, 0, stream>>>(zpart, NPIX / 256, fstats + 0);

    size_t woff = 0;
    for (int i = 0; i < 50; ++i) {
        const int cin = 1 + i;
        const int dil = (i % 10) + 1;
        const float* w  = sep ? (const float*)d_in[5 + i]   : lw0 + woff;
        const float* cb = sep ? (const float*)d_in[55 + i]  : lb0 + i;
        const float* g  = sep ? (const float*)d_in[105 + i] : lg0 + i;
        const float* be = sep ? (const float*)d_in[155 + i] : lt0 + i;
        conv_layer_kernel<<<NB * WD, 256, 0, stream>>>(feats, w, cb, ypart, cin, dil);
        bn_apply_kernel<<<NPIX / 256, 256, 0, stream>>>(feats, cin, ypart, g, be, zpart);
        finalize_kernel<<<1, 256, 0, stream>>>(zpart, NPIX / 256, fstats + 2 * cin);
        woff += (size_t)9 * cin;
    }

    bn2_prep_kernel<<<1, 64, 0, stream>>>(fstats, bn2_g, bn2_b, ST);
    final_conv_kernel<<<(NB * HW / 16) / 8, 256, 0, stream>>>(feats, ST, c2w, c2b, out);
}